// BlockSelfAttentionModule_4750233830016
// MI455X (gfx1250) — compile-verified
//
#include <hip/hip_runtime.h>

#define N_HEAD     8
#define NUM_INSTR  16
#define NUM_FRAMES 64
#define EMB        64
#define PAD_VAL    (-111.0f)
#define L_TOTAL    (NUM_FRAMES * NUM_INSTR)   // 1024
#define BH_TOTAL   32                          // B * N_HEAD

typedef __attribute__((ext_vector_type(2))) float v2f;
typedef __attribute__((ext_vector_type(4))) float v4f;
typedef __attribute__((ext_vector_type(8))) float v8f;

#define QROW 68   // padded row stride for Q tile in LDS (multiple of 4, avoids bank conflicts)
#define TROW 65   // padded row stride for Ut tile in LDS

__global__ __launch_bounds__(256)
void attn_fused_kernel(const float* __restrict__ q,       // (32, 1024, 64)
                       const float* __restrict__ r_inst,  // (16, 16, 64, 8)
                       const float* __restrict__ e_past,  // (64, 64, 8)
                       float* __restrict__ out)           // (32, 1024, 1024)
{
    __shared__ float Qs[16 * QROW];    // Q tile, row-padded
    __shared__ float Uts[16 * TROW];   // Ut tile [m][j]
    __shared__ float Uis[16 * 16];     // Ui tile [m][c]

    const int f   = blockIdx.x;        // frame 0..63
    const int bh  = blockIdx.y;        // 0..31
    const int h   = bh & (N_HEAD - 1);
    const int tid = threadIdx.x;       // 0..255

    // ---- Stage A: load contiguous 16x64 Q tile (1024 floats) into padded LDS ----
    {
        const v4f* src = (const v4f*)(q + ((size_t)bh * L_TOTAL + (size_t)f * NUM_INSTR) * EMB);
        const v4f v = src[tid];
        const int row = tid >> 4;      // 16 v4f per row
        const int seg = tid & 15;
        *(v4f*)(&Qs[row * QROW + seg * 4]) = v;
    }
    __syncthreads();

    if (tid < 128) {
        // ---- Stage B: Ut tile via fp32 WMMA. Wave w owns j-tile [16w, 16w+16). ----
        const int wave  = tid >> 5;            // 0..3
        const int lane  = tid & 31;
        const int mrow  = lane & 15;           // A: M row; B/D: N col
        const int khalf = (lane >> 4) * 2;     // K sub-offset: 0 (lanes 0-15) / 2 (lanes 16-31)

        const int jcol = wave * 16 + mrow;     // global j for B fragment
        const int r    = (f - jcol + 65) % 65; // relative-time row; 64 => PAD row

        v8f acc = {};
        const float* qrow = &Qs[mrow * QROW];
        #pragma unroll
        for (int k0 = 0; k0 < EMB; k0 += 4) {
            v2f a, b;
            a.x = qrow[k0 + khalf];
            a.y = qrow[k0 + khalf + 1];
            if (r == 64) {
                b.x = PAD_VAL; b.y = PAD_VAL;
            } else {
                const float* ep = e_past + ((size_t)r * EMB + (k0 + khalf)) * N_HEAD + h;
                b.x = ep[0];
                b.y = ep[N_HEAD];
            }
            // D = A(16x4 f32) * B(4x16 f32) + C   -> v_wmma_f32_16x16x4_f32
            acc = __builtin_amdgcn_wmma_f32_16x16x4_f32(
                      false, a, false, b, (short)0, acc, false, false);
        }
        // D layout: lane n = lane&15, VGPR v -> M = v + 8*(lane>=16)
        const int half8 = (lane >> 4) * 8;
        #pragma unroll
        for (int v = 0; v < 8; ++v) {
            Uts[(v + half8) * TROW + wave * 16 + mrow] = acc[v];
        }
    } else {
        // ---- Stage C: Ui tile via VALU (per-row distinct R matrix -> not GEMM-shaped) ----
        const int tt = tid - 128;              // 0..127, 2 outputs each
        #pragma unroll
        for (int rep = 0; rep < 2; ++rep) {
            const int o = tt + rep * 128;      // 0..255
            const int m = o >> 4;              // row == instrument i_l
            const int c = o & 15;              // target instrument
            const float* rp   = r_inst + ((size_t)(m * NUM_INSTR + c) * EMB) * N_HEAD + h;
            const float* qrow = &Qs[m * QROW];
            float s = 0.0f;
            #pragma unroll 8
            for (int d = 0; d < EMB; ++d) {
                s = fmaf(qrow[d], rp[(size_t)d * N_HEAD], s);
            }
            Uis[m * 16 + c] = s;
        }
    }
    __syncthreads();

    // ---- Stage D: stream 16x1024 output tile (64 KB), coalesced v4f NT stores ----
    const int m      = tid >> 4;               // row 0..15
    const int lane16 = tid & 15;
    float*       orow = out + ((size_t)bh * L_TOTAL + (size_t)f * NUM_INSTR + m) * L_TOTAL;
    const float* utr  = &Uts[m * TROW];
    const float* uir  = &Uis[m * 16];
    #pragma unroll
    for (int g = 0; g < 16; ++g) {
        const int cb = (g * 16 + lane16) * 4;  // column base, 0..1020
        const int fm = cb >> 4;                // frame of column
        const int im = cb & 15;                // instrument base of column (multiple of 4)
        const float ut = utr[fm];
        v4f val;
        val.x = ut + uir[im + 0];
        val.y = ut + uir[im + 1];
        val.z = ut + uir[im + 2];
        val.w = ut + uir[im + 3];
        __builtin_nontemporal_store(val, (v4f*)(orow + cb));
    }
}

extern "C" void kernel_launch(void* const* d_in, const int* in_sizes, int n_in,
                              void* d_out, int out_size, void* d_ws, size_t ws_size,
                              hipStream_t stream) {
    const float* q = (const float*)d_in[0];   // (32, 1024, 64)
    const float* r = (const float*)d_in[1];   // (16, 16, 64, 8)
    const float* e = (const float*)d_in[2];   // (64, 64, 8)
    // d_in[3] = flipped_masks: unused by the reference computation.
    float* out = (float*)d_out;               // (32, 1024, 1024)

    dim3 grid(NUM_FRAMES, BH_TOTAL);          // 64 x 32 = 2048 blocks
    attn_fused_kernel<<<grid, 256, 0, stream>>>(q, r, e, out);
}